// IMLE_29351806501523
// MI455X (gfx1250) — compile-verified
//
#include <hip/hip_runtime.h>
#include <math.h>
#include <stdint.h>

// Problem constants (from reference setup_inputs)
#define B_ 16
#define M_ 32
#define K_ 20
#define A_ 8
#define T_ 30
// D = 2, unnormalize(x) = 50*x  (FUT_MIN=-50, FUT_MAX=50)

// Padded LDS row stride (floats) for one target k2 row: 480 data + 4 pad.
// 484*4 = 1936 bytes -> 16B aligned rows; bank stride 484%64 = 36 -> nearly
// conflict-free across the 20 k2 rows for b64 reads.
#define TGT_ROW 484

__device__ __forceinline__ unsigned lds_off_of(const void* p) {
  // Generic pointer to LDS: low 32 bits are the wave-relative LDS byte offset.
  return (unsigned)(uintptr_t)p;
}

// CDNA5 async global->LDS copy (ASYNCcnt-tracked, no VGPR round trip).
// GVS mode: addr = SGPR64 base + VGPR32 byte offset. VDST VGPR = LDS byte addr.
__device__ __forceinline__ void async_copy_b128(unsigned lds_byte_off,
                                                const float* base,
                                                unsigned voff_bytes) {
  asm volatile("global_load_async_to_lds_b128 %0, %1, %2"
               :: "v"(lds_byte_off), "v"(voff_bytes), "s"(base)
               : "memory");
}

__device__ __forceinline__ void wait_async_zero() {
  asm volatile("s_wait_asynccnt 0" ::: "memory");
}

// One workgroup per (b, m). Computes chamfer[b,m] and nn_gt[b,m].
__launch_bounds__(256)
__global__ void imle_bm_kernel(const float* __restrict__ imle_gen,
                               const float* __restrict__ target,
                               const float* __restrict__ gt,
                               float* __restrict__ ws_chamfer,
                               float* __restrict__ ws_nngt) {
  __shared__ float sh_tgt[K_ * TGT_ROW];   // 9680 f  (padded k2 rows)
  __shared__ float sh_gt[A_ * T_ * 2];     //  480 f
  __shared__ float sh_pd[A_ * K_ * K_];    // 3200 f  pd[a][k1][k2]
  __shared__ float sh_d[A_ * K_];          //  160 f  d[a][k]
  __shared__ float sh_red[256];
  __shared__ float sh_min[A_];

  const int tid = threadIdx.x;
  const int bid = blockIdx.x;              // b*M + m
  const int b   = bid / M_;

  const float* gen_b = imle_gen + (size_t)bid * (K_ * A_ * T_ * 2);
  const float* tgt_b = target   + (size_t)b   * (K_ * A_ * T_ * 2);
  const float* gt_b  = gt       + (size_t)b   * (A_ * T_ * 2);

  // ---- stage target tile (row-padded) and gt tile into LDS via async DMA ----
  for (int c = tid; c < K_ * 120; c += 256) {       // 120 x b128 chunks per k2 row
    const int k2 = c / 120, j = c % 120;
    async_copy_b128(lds_off_of(&sh_tgt[k2 * TGT_ROW + j * 4]), tgt_b,
                    (unsigned)((k2 * 480 + j * 4) * 4));
  }
  for (int c = tid; c < (A_ * T_ * 2) / 4; c += 256) {  // 120 chunks
    async_copy_b128(lds_off_of(&sh_gt[c * 4]), gt_b, (unsigned)(c * 16));
  }

  // ---- warm L0/L2 for this block's 38.4 KB gen stream while the DMA runs ----
  // global_prefetch_b8: no VGPR return, no counter traffic; whole cacheline.
  for (int off = tid * 128; off < K_ * A_ * T_ * 2 * 4; off += 256 * 128) {
    __builtin_prefetch((const char*)gen_b + off, 0, 3);
  }

  wait_async_zero();     // this wave's async copies landed in LDS
  __syncthreads();       // all waves' copies visible

  // ---- pairwise distances pd[a][k1][k2] = (50/T) * sum_t sqrt(dx^2+dy^2) ----
  // Lane order: k2 fastest -> gen row broadcast across 20 lanes; tgt LDS reads
  // stride TGT_ROW across lanes (conflict-light).
  const float cscale = 50.0f / (float)T_;
  for (int e = tid; e < A_ * K_ * K_; e += 256) {
    const int k2 = e % K_;
    const int r  = e / K_;
    const int k1 = r % K_;
    const int a  = r / K_;
    const float* __restrict__ gp = gen_b + ((k1 * A_ + a) * T_) * 2;     // global (cached)
    const float* __restrict__ tp = &sh_tgt[k2 * TGT_ROW + (a * T_) * 2]; // LDS
    float s = 0.0f;
#pragma unroll
    for (int t = 0; t < T_; ++t) {
      const float dx = gp[2 * t]     - tp[2 * t];
      const float dy = gp[2 * t + 1] - tp[2 * t + 1];
      s += __builtin_amdgcn_sqrtf(fmaf(dx, dx, dy * dy));
    }
    sh_pd[(a * K_ + k1) * K_ + k2] = s * cscale;
  }

  // ---- gt term d[a][k] = (1/T) * sum_t ||50*gen - gt|| ----
  for (int e = tid; e < A_ * K_; e += 256) {
    const int k = e % K_;
    const int a = e / K_;
    const float* __restrict__ gp = gen_b + ((k * A_ + a) * T_) * 2;
    const float* __restrict__ gq = &sh_gt[(a * T_) * 2];
    float s = 0.0f;
#pragma unroll
    for (int t = 0; t < T_; ++t) {
      const float dx = fmaf(50.0f, gp[2 * t],     -gq[2 * t]);
      const float dy = fmaf(50.0f, gp[2 * t + 1], -gq[2 * t + 1]);
      s += __builtin_amdgcn_sqrtf(fmaf(dx, dx, dy * dy));
    }
    sh_d[a * K_ + k] = s * (1.0f / (float)T_);
  }
  __syncthreads();

  // ---- chamfer reductions: row-mins (g2t) + col-mins (t2g), then tree-sum ----
  float partial = 0.0f;
  if (tid < A_ * K_) {
    const int a = tid / K_, i = tid % K_;
    const float* row = &sh_pd[(a * K_ + i) * K_];   // fixed k1=i, scan k2
    float mr = row[0];
#pragma unroll
    for (int k2 = 1; k2 < K_; ++k2) mr = fminf(mr, row[k2]);
    const float* col = &sh_pd[a * K_ * K_ + i];     // fixed k2=i, scan k1
    float mc = col[0];
#pragma unroll
    for (int k1 = 1; k1 < K_; ++k1) mc = fminf(mc, col[k1 * K_]);
    partial = mr + mc;
  }
  sh_red[tid] = partial;
  __syncthreads();
#pragma unroll
  for (int off = 128; off > 0; off >>= 1) {
    if (tid < off) sh_red[tid] += sh_red[tid + off];
    __syncthreads();
  }

  // ---- gt-term min over K per agent ----
  if (tid < A_) {
    float mn = sh_d[tid * K_];
#pragma unroll
    for (int k = 1; k < K_; ++k) mn = fminf(mn, sh_d[tid * K_ + k]);
    sh_min[tid] = mn;
  }
  __syncthreads();

  if (tid == 0) {
    ws_chamfer[bid] = sh_red[0] * (1.0f / (float)(A_ * K_));
    float nn = 0.0f;
#pragma unroll
    for (int a = 0; a < A_; ++a) nn += sh_min[a];
    ws_nngt[bid] = nn * (1.0f / (float)A_);
  }
}

// min over M per batch, mean over B, emit [loss, loss_chamfer, loss_gt]
__global__ void imle_finalize(const float* __restrict__ wc,
                              const float* __restrict__ wn,
                              float* __restrict__ out) {
  __shared__ float sc[B_], sn[B_];
  const int t = threadIdx.x;
  if (t < B_) {
    float mc = wc[t * M_], mn = wn[t * M_];
    for (int m = 1; m < M_; ++m) {
      mc = fminf(mc, wc[t * M_ + m]);
      mn = fminf(mn, wn[t * M_ + m]);
    }
    sc[t] = mc; sn[t] = mn;
  }
  __syncthreads();
  if (t == 0) {
    float c = 0.0f, g = 0.0f;
    for (int bb = 0; bb < B_; ++bb) { c += sc[bb]; g += sn[bb]; }
    c *= (1.0f / (float)B_);
    g *= (1.0f / (float)B_);
    out[0] = c + g;   // loss (CHAMFER_W = GT_W = 1)
    out[1] = c;       // loss_chamfer
    out[2] = g;       // loss_gt
  }
}

extern "C" void kernel_launch(void* const* d_in, const int* in_sizes, int n_in,
                              void* d_out, int out_size, void* d_ws, size_t ws_size,
                              hipStream_t stream) {
  const float* imle_gen = (const float*)d_in[0];  // (B,M,K,A,T,2)
  const float* target   = (const float*)d_in[1];  // (B,K,A,T,2)
  const float* gt       = (const float*)d_in[2];  // (B,A,T,2)

  float* wc = (float*)d_ws;            // B*M chamfer values
  float* wn = wc + (B_ * M_);          // B*M nn_gt values

  imle_bm_kernel<<<B_ * M_, 256, 0, stream>>>(imle_gen, target, gt, wc, wn);
  imle_finalize<<<1, 32, 0, stream>>>(wc, wn, (float*)d_out);

  (void)in_sizes; (void)n_in; (void)out_size; (void)ws_size;
}